// CriticNetwork_66314295050520
// MI455X (gfx1250) — compile-verified
//
#include <hip/hip_runtime.h>

typedef __attribute__((ext_vector_type(2))) float v2f;
typedef __attribute__((ext_vector_type(8))) float v8f;

// Guaranteed-native non-returning fp32 global atomic add (device scope).
__device__ __forceinline__ void atomic_add_f32_dev(float* p, float v) {
  asm volatile("global_atomic_add_f32 %0, %1, off scope:SCOPE_DEV"
               :: "v"(p), "v"(v) : "memory");
}

// Four adds sharing one VGPR-pair address, folded into instruction offsets.
__device__ __forceinline__ void atomic_add_f32x4_dev(float* p, float4 v) {
  asm volatile(
      "global_atomic_add_f32 %0, %1, off scope:SCOPE_DEV\n\t"
      "global_atomic_add_f32 %0, %2, off offset:4 scope:SCOPE_DEV\n\t"
      "global_atomic_add_f32 %0, %3, off offset:8 scope:SCOPE_DEV\n\t"
      "global_atomic_add_f32 %0, %4, off offset:12 scope:SCOPE_DEV"
      :: "v"(p), "v"(v.x), "v"(v.y), "v"(v.z), "v"(v.w) : "memory");
}

// ---------------------------------------------------------------------------
// 0) Re-initialize scratch every call: agg=0, deg=1 (self-loop), v=0.
// ---------------------------------------------------------------------------
__global__ void init_ws_k(float* __restrict__ agg, float* __restrict__ deg,
                          float* __restrict__ v, int N) {
  size_t i = (size_t)blockIdx.x * blockDim.x + threadIdx.x;
  size_t tot = (size_t)N * 128;
  if (i < tot) agg[i] = 0.0f;
  if (i < (size_t)N) deg[i] = 1.0f;   // +1 from the self-loop on every node
  if (i < 128) v[i] = 0.0f;
}

// ---------------------------------------------------------------------------
// 1) Degree count over real edges (self-loop already in init), then rsqrt.
// ---------------------------------------------------------------------------
__global__ void degree_k(const int* __restrict__ dst, float* __restrict__ deg, int E) {
  int e = blockIdx.x * blockDim.x + threadIdx.x;
  if (e < E) atomic_add_f32_dev(&deg[dst[e]], 1.0f);
}

__global__ void dinv_k(float* __restrict__ deg, int N) {
  int n = blockIdx.x * blockDim.x + threadIdx.x;
  if (n < N) deg[n] = rsqrtf(deg[n]);   // deg >= 1 always, in-place -> dinv
}

// ---------------------------------------------------------------------------
// 2) xl = x @ Wg.T  via V_WMMA_F32_16X16X4_F32.
//    One wave32 per 16-row tile (N = 50000 = 16*3125, exact).
//    A-layout (16x4): lane(half,m): comp0=K(kk+2*half), comp1=K(kk+2*half+1), M=m
//    B-layout (4x16): symmetric, N=m, B[k][n] = Wg[n][k] (contiguous float2)
//    D-layout (16x16): acc[i] -> M = i + 8*half, N = m
// ---------------------------------------------------------------------------
__global__ void __launch_bounds__(32)
gemm_xWgT_k(const float* __restrict__ x, const float* __restrict__ Wg,
            float* __restrict__ xl) {
  const int lane = threadIdx.x;      // 0..31 (wave32)
  const int half = lane >> 4;        // 0 | 1
  const int m    = lane & 15;
  const int row0 = blockIdx.x * 16;

  const v8f zero = {0.f, 0.f, 0.f, 0.f, 0.f, 0.f, 0.f, 0.f};
  v8f acc[8];
#pragma unroll
  for (int t = 0; t < 8; ++t) acc[t] = zero;

  for (int kk = 0; kk < 128; kk += 4) {
    const int k0 = kk + half * 2;
    v2f a = *(const v2f*)(x + (size_t)(row0 + m) * 128 + k0);
#pragma unroll
    for (int t = 0; t < 8; ++t) {
      v2f b = *(const v2f*)(Wg + (size_t)(t * 16 + m) * 128 + k0);
      acc[t] = __builtin_amdgcn_wmma_f32_16x16x4_f32(
          /*neg_a=*/false, a, /*neg_b=*/false, b,
          /*c_mod=*/(short)0, acc[t], /*reuse_a=*/false, /*reuse_b=*/false);
    }
  }

#pragma unroll
  for (int t = 0; t < 8; ++t)
#pragma unroll
    for (int i = 0; i < 8; ++i)
      xl[(size_t)(row0 + half * 8 + i) * 128 + t * 16 + m] = acc[t][i];
}

// ---------------------------------------------------------------------------
// 3) Edge scatter: one wave per edge; 4 floats/lane (float4 gather, L2-resident),
//    4 native non-returning fp32 atomics per lane into agg[dst].
// ---------------------------------------------------------------------------
__global__ void edge_scatter_k(const float* __restrict__ xl, const int* __restrict__ src,
                               const int* __restrict__ dst, const float* __restrict__ dinv,
                               float* __restrict__ agg, int E) {
  int e = (int)(((size_t)blockIdx.x * blockDim.x + threadIdx.x) >> 5);
  int lane = threadIdx.x & 31;
  if (e >= E) return;
  int s = src[e];
  int d = dst[e];
  float norm = dinv[s] * dinv[d];
  float4 mv = *reinterpret_cast<const float4*>(xl + (size_t)s * 128 + lane * 4);
  mv.x *= norm; mv.y *= norm; mv.z *= norm; mv.w *= norm;
  atomic_add_f32x4_dev(agg + (size_t)d * 128 + lane * 4, mv);
}

// ---------------------------------------------------------------------------
// 4) Finalize: self-loop term + bias, ReLU, residual, sum-pool into v[128].
//    Block = 2 rows x 128 dims; grid-stride; one atomic per dim per block.
// ---------------------------------------------------------------------------
__global__ void __launch_bounds__(256)
finalize_pool_k(const float* __restrict__ agg, const float* __restrict__ xl,
                const float* __restrict__ x, const float* __restrict__ dinv,
                const float* __restrict__ bg, float* __restrict__ v, int N) {
  __shared__ float sv[256];
  const int d = threadIdx.x & 127;
  const int r = threadIdx.x >> 7;     // 0 | 1
  float accum = 0.0f;
  const float bias = bg[d];
  for (int n = blockIdx.x * 2 + r; n < N; n += gridDim.x * 2) {
    float di = dinv[n];
    float h = agg[(size_t)n * 128 + d] + xl[(size_t)n * 128 + d] * di * di + bias;
    h = fmaxf(h, 0.0f) + x[(size_t)n * 128 + d];
    accum += h;
  }
  sv[threadIdx.x] = accum;
  __syncthreads();
  if (threadIdx.x < 128)
    atomic_add_f32_dev(&v[d], sv[threadIdx.x] + sv[threadIdx.x + 128]);
}

// ---------------------------------------------------------------------------
// 5) MLP head, single block of 256 threads (0.3 MFLOP -> negligible).
// ---------------------------------------------------------------------------
__global__ void __launch_bounds__(256)
mlp_head_k(const float* __restrict__ v,
           const float* __restrict__ W1, const float* __restrict__ b1,
           const float* __restrict__ W2, const float* __restrict__ b2,
           const float* __restrict__ W3, const float* __restrict__ b3,
           float* __restrict__ out) {
  __shared__ float sv[128];
  __shared__ float h1[512];
  __shared__ float h2[256];
  __shared__ float red[256];
  const int t = threadIdx.x;

  if (t < 128) sv[t] = v[t];
  __syncthreads();

  for (int r = t; r < 512; r += 256) {
    float acc = b1[r];
    const float* w = W1 + (size_t)r * 128;
    for (int k = 0; k < 128; ++k) acc = fmaf(w[k], sv[k], acc);
    h1[r] = fmaxf(acc, 0.0f);
  }
  __syncthreads();

  {
    float acc = b2[t];
    const float* w = W2 + (size_t)t * 512;
    for (int k = 0; k < 512; ++k) acc = fmaf(w[k], h1[k], acc);
    h2[t] = fmaxf(acc, 0.0f);
  }
  __syncthreads();

  red[t] = W3[t] * h2[t];
  __syncthreads();
  for (int s = 128; s > 0; s >>= 1) {
    if (t < s) red[t] += red[t + s];
    __syncthreads();
  }
  if (t == 0) out[0] = red[0] + b3[0];
}

// ---------------------------------------------------------------------------
extern "C" void kernel_launch(void* const* d_in, const int* in_sizes, int n_in,
                              void* d_out, int out_size, void* d_ws, size_t ws_size,
                              hipStream_t stream) {
  const float* x  = (const float*)d_in[0];
  const int*   ei = (const int*)d_in[1];
  const float* Wg = (const float*)d_in[2];
  const float* bg = (const float*)d_in[3];
  const float* W1 = (const float*)d_in[4];
  const float* b1 = (const float*)d_in[5];
  const float* W2 = (const float*)d_in[6];
  const float* b2 = (const float*)d_in[7];
  const float* W3 = (const float*)d_in[8];
  const float* b3 = (const float*)d_in[9];

  const int D = 128;
  const int N = in_sizes[0] / D;   // 50000 (multiple of 16)
  const int E = in_sizes[1] / 2;   // 800000
  const int* src = ei;
  const int* dst = ei + E;

  // Workspace layout (floats): xl[N*128] | agg[N*128] | deg->dinv[N] | v[128]
  float* xl  = (float*)d_ws;
  float* agg = xl + (size_t)N * D;
  float* deg = agg + (size_t)N * D;
  float* v   = deg + N;

  {
    int blocks = (int)(((size_t)N * D + 255) / 256);
    init_ws_k<<<blocks, 256, 0, stream>>>(agg, deg, v, N);
  }
  degree_k<<<(E + 255) / 256, 256, 0, stream>>>(dst, deg, E);
  dinv_k<<<(N + 255) / 256, 256, 0, stream>>>(deg, N);
  gemm_xWgT_k<<<N / 16, 32, 0, stream>>>(x, Wg, xl);
  edge_scatter_k<<<(E + 7) / 8, 256, 0, stream>>>(xl, src, dst, deg, agg, E);
  finalize_pool_k<<<512, 256, 0, stream>>>(agg, xl, x, deg, bg, v, N);
  mlp_head_k<<<1, 256, 0, stream>>>(v, W1, b1, W2, b2, W3, b3, (float*)d_out);
}